// CKConv_24378234372295
// MI455X (gfx1250) — compile-verified
//
#include <hip/hip_runtime.h>

// CKConv on MI455X (gfx1250): FFT-based causal convolution.
// - 16384-pt complex f32 Stockham FFT entirely in one WGP's LDS (2x128KB
//   ping-pong <= 320 KB CDNA5 LDS), 1024 threads/block for latency hiding.
// - Global<->LDS moved with gfx1250 async-to-LDS ops (ASYNCcnt-tracked,
//   VGPR-bypassing) -- builtin signature: (v4i AS1*, v4i AS3*, imm, imm).
// - WMMA f32 16x16x4 for the SIREN output GEMM (w3@h2, M=1024 N=8192 K=32),
//   fused with the causal flip / zero-pad / complex-pack epilogue.
//
// Constants: B=8, L=8192, Ci=Co=H=32, n=16384.
// Workspace (floats): h2 262,144 | Kc 33,554,432 | Xc 8,388,608 | Yc 8,388,608
// (~202.4 MB of d_ws).

#define LL    8192
#define NN    16384
#define HH    32
#define CI    32
#define CO    32
#define BB    8
#define OMEGA 32.5f

typedef __attribute__((ext_vector_type(2))) float v2f;
typedef __attribute__((ext_vector_type(8))) float v8f;

#if defined(__has_builtin)
#if __has_builtin(__builtin_amdgcn_global_load_async_to_lds_b128) && \
    __has_builtin(__builtin_amdgcn_global_store_async_from_lds_b128) && \
    __has_builtin(__builtin_amdgcn_s_wait_asynccnt)
#define USE_ASYNC_LDS 1
#endif
#endif
#ifndef USE_ASYNC_LDS
#define USE_ASYNC_LDS 0
#endif

#if USE_ASYNC_LDS
typedef int v4i_base __attribute__((ext_vector_type(4)));
typedef __attribute__((address_space(1))) v4i_base gas_v4i;  // global b128
typedef __attribute__((address_space(3))) v4i_base las_v4i;  // LDS b128
#endif

// ---------------------------------------------------------------- zero fill
__global__ void fill_zero_f4(float4* __restrict__ p, unsigned long long n4) {
    unsigned long long g = (unsigned long long)blockIdx.x * blockDim.x + threadIdx.x;
    if (g < n4) p[g] = make_float4(0.f, 0.f, 0.f, 0.f);
}

// ------------------------------------------------- SIREN layers 1+2 (VALU)
__global__ void siren_h2(const float* __restrict__ w1, const float* __restrict__ b1,
                         const float* __restrict__ w2, const float* __restrict__ b2,
                         float* __restrict__ h2) {
    int l = blockIdx.x * blockDim.x + threadIdx.x;   // 8192 threads
    if (l >= LL) return;
    float t = -1.0f + 2.0f * (float)l / (float)(LL - 1);
    float h1[HH];
#pragma unroll
    for (int j = 0; j < HH; ++j) h1[j] = __sinf(OMEGA * (w1[j] * t + b1[j]));
#pragma unroll 4
    for (int j = 0; j < HH; ++j) {
        float s = b2[j];
#pragma unroll
        for (int m = 0; m < HH; ++m) s += w2[j * HH + m] * h1[m];
        h2[(size_t)j * LL + l] = __sinf(OMEGA * s);
    }
}

// ---------------------------------- k = w3 @ h2 + b3  via V_WMMA_F32_16X16X4
// One wave per 16x16 tile; epilogue writes the flipped, left-padded kernel
// directly as the complex FFT input: kf[t] = k[8191-t], tail pre-zeroed.
__global__ void k_gemm_wmma(const float* __restrict__ w3, const float* __restrict__ b3,
                            const float* __restrict__ h2, float2* __restrict__ Kc) {
    const int wave = threadIdx.x >> 5;             // 4 waves / block
    const int lane = threadIdx.x & 31;
    const int tile = blockIdx.x * 4 + wave;        // 32768 tiles total
    const int tileM = tile >> 9;
    const int tileN = tile & 511;
    const int row0 = tileM << 4;
    const int col0 = tileN << 4;
    const int half = lane >> 4;
    const int l16  = lane & 15;

    v8f acc = {};
#pragma unroll
    for (int k0 = 0; k0 < HH; k0 += 4) {
        v2f a, b;
        a[0] = w3[(size_t)(row0 + l16) * HH + k0 + 0 + 2 * half];
        a[1] = w3[(size_t)(row0 + l16) * HH + k0 + 1 + 2 * half];
        b[0] = h2[(size_t)(k0 + 0 + 2 * half) * LL + col0 + l16];
        b[1] = h2[(size_t)(k0 + 1 + 2 * half) * LL + col0 + l16];
        acc = __builtin_amdgcn_wmma_f32_16x16x4_f32(false, a, false, b,
                                                    (short)0, acc, false, false);
    }
#pragma unroll
    for (int e = 0; e < 8; ++e) {
        int r = row0 + e + 8 * half;               // flat (o*Ci + i)
        int l = col0 + l16;
        float v = acc[e] + b3[r];
        Kc[(size_t)r * NN + (LL - 1 - l)] = make_float2(v, 0.0f);
    }
}

// -------------------------------------------------- pack x -> complex, padded
__global__ void pack_x(const float* __restrict__ x, float2* __restrict__ Xc) {
    int g = blockIdx.x * blockDim.x + threadIdx.x;   // B*Ci*L
    int t   = g & (LL - 1);
    int sig = g >> 13;                                // b*Ci + i
    int b = sig >> 5, i = sig & 31;
    Xc[(size_t)sig * NN + t] =
        make_float2(x[((size_t)b * LL + t) * CI + i], 0.0f);
}

// ------------------------------ 16384-pt complex Stockham FFT, all in LDS
// One workgroup (1024 thr = 32 waves) per signal; 256 KB dynamic LDS.
// sign = -1 forward, +1 inverse (unscaled; 1/n applied in emit).
__global__ void fft16k(float2* __restrict__ data, int sign) {
    extern __shared__ float2 lds[];
    float2* A = lds;
    float2* Bb = lds + NN;
    const size_t base = (size_t)blockIdx.x * NN;

#if USE_ASYNC_LDS
    // async global->LDS, 16B per op, bypasses VGPRs (ASYNCcnt-tracked)
    for (int t = threadIdx.x * 2; t < NN; t += blockDim.x * 2) {
        __builtin_amdgcn_global_load_async_to_lds_b128(
            (gas_v4i*)(data + base + t), (las_v4i*)(A + t), 0, 0);
    }
    __builtin_amdgcn_s_wait_asynccnt(0);
#else
    for (int t = threadIdx.x; t < NN; t += blockDim.x) A[t] = data[base + t];
#endif
    __syncthreads();

    float2* src = A;
    float2* dst = Bb;
    const float sgn = (float)sign;
    int sh = 0;
    for (int half = NN >> 1; half >= 1; half >>= 1, ++sh) {   // 14 stages
        const int s = 1 << sh;
        const float theta = sgn * 3.14159265358979323846f / (float)half;
        for (int idx = threadIdx.x; idx < (NN >> 1); idx += blockDim.x) {
            const int p = idx >> sh;
            const int q = idx & (s - 1);
            const float2 a = src[q + s * p];
            const float2 b = src[q + s * (p + half)];
            float sn, cs;
            __sincosf(theta * (float)p, &sn, &cs);
            const float tr = a.x - b.x, ti = a.y - b.y;
            dst[q + s * (2 * p)]     = make_float2(a.x + b.x, a.y + b.y);
            dst[q + s * (2 * p + 1)] = make_float2(tr * cs - ti * sn,
                                                   tr * sn + ti * cs);
        }
        __syncthreads();
        float2* tmp = src; src = dst; dst = tmp;
    }
    // 14 swaps (even) -> result back in A == src
#if USE_ASYNC_LDS
    for (int t = threadIdx.x * 2; t < NN; t += blockDim.x * 2) {
        __builtin_amdgcn_global_store_async_from_lds_b128(
            (gas_v4i*)(data + base + t), (las_v4i*)(src + t), 0, 0);
    }
    __builtin_amdgcn_s_wait_asynccnt(0);
#else
    for (int t = threadIdx.x; t < NN; t += blockDim.x) data[base + t] = src[t];
#endif
}

// --------------------------- frequency-domain contraction over Ci (BW-bound)
__global__ void freq_mul(const float2* __restrict__ Xc, const float2* __restrict__ Kc,
                         float2* __restrict__ Yc) {
    size_t g = (size_t)blockIdx.x * blockDim.x + threadIdx.x; // B*Co*N
    int f    = (int)(g & (NN - 1));
    int pair = (int)(g >> 14);                                // b*Co + o
    int b = pair >> 5, o = pair & 31;
    float re = 0.f, im = 0.f;
#pragma unroll 4
    for (int i = 0; i < CI; ++i) {
        float2 xv = Xc[((size_t)(b * CI + i)) * NN + f];
        float2 kv = Kc[((size_t)(o * CI + i)) * NN + f];
        re += xv.x * kv.x - xv.y * kv.y;
        im += xv.x * kv.y + xv.y * kv.x;
    }
    Yc[(size_t)pair * NN + f] = make_float2(re, im);
}

// ------------------------------ emit: 1/n scale, + bias, [B,Co,L] -> [B,L,Co]
__global__ void emit_out(const float2* __restrict__ Yc, const float* __restrict__ bias,
                         float* __restrict__ out) {
    size_t g = (size_t)blockIdx.x * blockDim.x + threadIdx.x; // B*L*Co
    int o = (int)(g & 31);
    int l = (int)((g >> 5) & (LL - 1));
    int b = (int)(g >> 18);
    float v = Yc[((size_t)(b * CO + o)) * NN + l].x * (1.0f / (float)NN) + bias[o];
    out[((size_t)b * LL + l) * CO + o] = v;
}

extern "C" void kernel_launch(void* const* d_in, const int* in_sizes, int n_in,
                              void* d_out, int out_size, void* d_ws, size_t ws_size,
                              hipStream_t stream) {
    (void)in_sizes; (void)n_in; (void)out_size; (void)ws_size;
    const float* x    = (const float*)d_in[0];
    const float* w1   = (const float*)d_in[1];
    const float* b1   = (const float*)d_in[2];
    const float* w2   = (const float*)d_in[3];
    const float* b2   = (const float*)d_in[4];
    const float* w3   = (const float*)d_in[5];
    const float* b3   = (const float*)d_in[6];
    const float* bias = (const float*)d_in[7];
    float* out = (float*)d_out;

    float*  ws = (float*)d_ws;
    float*  h2 = ws;                                           //   262,144 f
    float2* Kc = (float2*)(ws + 262144);                       // 33,554,432 f
    float2* Xc = (float2*)(ws + 262144 + 33554432);            //  8,388,608 f
    float2* Yc = (float2*)(ws + 262144 + 33554432 + 8388608);  //  8,388,608 f

    // 256 KB dynamic LDS for the in-WGP FFT (CDNA5: up to 320 KB / workgroup)
    (void)hipFuncSetAttribute(reinterpret_cast<const void*>(fft16k),
                              hipFuncAttributeMaxDynamicSharedMemorySize, 2 * NN * 8);

    // zero Kc + Xc (contiguous): tails of the padded FFT inputs
    {
        unsigned long long n4 = (33554432ull + 8388608ull) / 4ull;
        fill_zero_f4<<<(unsigned)((n4 + 255) / 256), 256, 0, stream>>>((float4*)Kc, n4);
    }
    siren_h2<<<LL / 256, 256, 0, stream>>>(w1, b1, w2, b2, h2);
    k_gemm_wmma<<<8192, 128, 0, stream>>>(w3, b3, h2, Kc);     // 32768 WMMA tiles
    pack_x<<<(BB * CI * LL) / 256, 256, 0, stream>>>(x, Xc);

    fft16k<<<CO * CI, 1024, 2 * NN * 8, stream>>>(Kc, -1);     // 1024 kernel FFTs
    fft16k<<<BB * CI, 1024, 2 * NN * 8, stream>>>(Xc, -1);     //  256 input  FFTs
    freq_mul<<<(BB * CO * NN) / 256, 256, 0, stream>>>(Xc, Kc, Yc);
    fft16k<<<BB * CO, 1024, 2 * NN * 8, stream>>>(Yc, +1);     //  256 inverse FFTs

    emit_out<<<(BB * LL * CO) / 256, 256, 0, stream>>>(Yc, bias, out);
}